// YOLOLoss_72730976191060
// MI455X (gfx1250) — compile-verified
//
#include <hip/hip_runtime.h>
#include <hip/hip_bf16.h>
#include <math.h>

#define B_IMGS  16
#define N_PREDS 19200
#define T_TGT   32
#define C_CLS   80
#define BPI     16     // stage-1 blocks per image (256 WGs total)
#define BLK     256

typedef float v2f __attribute__((ext_vector_type(2)));
typedef float v8f __attribute__((ext_vector_type(8)));

__device__ __forceinline__ float softplusf(float x) {
    // matches jnp.logaddexp(0, x) = max(x,0) + log1p(exp(-|x|))
    return fmaxf(x, 0.0f) + log1pf(expf(-fabsf(x)));
}

// Exact f32 sum across the 32 lanes of a wave, routed through the WMMA pipe.
// A = ones(16x4), B holds the 32 lane values in its first B-operand VGPR.
// D[m][n] = sum_k B[k][n] for every m, so summing D over all 32 lanes counts
// every B entry exactly twice (two identical M-rows) -> total = 0.5 * lanesum.
// This is invariant to the exact (lane,vgpr)->(k,n) mapping of the B operand.
__device__ __forceinline__ float wave_sum32(float v) {
#if __has_builtin(__builtin_amdgcn_wmma_f32_16x16x4_f32)
    v2f a;  a.x = 1.0f; a.y = 1.0f;   // A: all ones
    v2f bm; bm.x = v;   bm.y = 0.0f;  // B: one value per lane, second slot zero
    v8f c = {0.0f, 0.0f, 0.0f, 0.0f, 0.0f, 0.0f, 0.0f, 0.0f};
    c = __builtin_amdgcn_wmma_f32_16x16x4_f32(false, a, false, bm,
                                              (short)0, c, false, false);
    float s = c[0];
    s += __shfl_xor(s, 1, 32);
    s += __shfl_xor(s, 2, 32);
    s += __shfl_xor(s, 4, 32);
    s += __shfl_xor(s, 8, 32);
    s += __shfl_xor(s, 16, 32);
    return s * 0.5f;
#else
    float s = v;
    for (int off = 16; off; off >>= 1) s += __shfl_xor(s, off, 32);
    return s;
#endif
}

// ---------------------------------------------------------------------------
// Stage 0: zero the argmax key slots (workspace is poisoned by the harness).
// ---------------------------------------------------------------------------
__global__ void yolo_init(unsigned long long* __restrict__ wkey) {
    int i = blockIdx.x * blockDim.x + threadIdx.x;
    if (i < B_IMGS * T_TGT) wkey[i] = 0ull;
}

// ---------------------------------------------------------------------------
// Stage 1: per (image, pred): IoU vs all 32 targets -> packed-max argmax,
// plus softplus(conf) partial sums. Key = (iou_bits << 32) | (~n) so that
// u64 max == (max iou, then min pred index) == jnp.argmax first-occurrence.
// ---------------------------------------------------------------------------
__global__ __launch_bounds__(BLK) void yolo_stage1(
    const float4* __restrict__ pboxes, const float* __restrict__ pconf,
    const float4* __restrict__ tboxes,
    unsigned long long* __restrict__ wkey, float* __restrict__ wsp) {

    __shared__ float4 sT[T_TGT];
    __shared__ float  sA2[T_TGT];
    __shared__ unsigned long long sKey[T_TGT];
    __shared__ float  sRed[BLK / 32];

    const int b   = blockIdx.x / BPI;
    const int blk = blockIdx.x % BPI;
    const int tid = threadIdx.x;
    const int STRIDE = BPI * BLK;

    if (tid < T_TGT) {
        float4 t = tboxes[b * T_TGT + tid];
        sT[tid]  = t;
        sA2[tid] = (t.z - t.x) * (t.w - t.y);
        sKey[tid] = 0ull;
    }
    __syncthreads();

    unsigned long long best[T_TGT];
#pragma unroll
    for (int t = 0; t < T_TGT; ++t) best[t] = 0ull;

    float spsum = 0.0f;
    for (int n = blk * BLK + tid; n < N_PREDS; n += STRIDE) {
        // gfx1250 prefetch of this thread's next grid-stride elements
        if (n + STRIDE < N_PREDS) {
            __builtin_prefetch(&pboxes[b * N_PREDS + n + STRIDE], 0, 0);
            __builtin_prefetch(&pconf [b * N_PREDS + n + STRIDE], 0, 0);
        }
        float4 p = pboxes[b * N_PREDS + n];
        float a1 = (p.z - p.x) * (p.w - p.y);
        spsum += softplusf(pconf[b * N_PREDS + n]);
        unsigned long long lo =
            (unsigned long long)(0xFFFFFFFFu - (unsigned)n);
#pragma unroll
        for (int t = 0; t < T_TGT; ++t) {
            float4 tb = sT[t];
            float ix1 = fmaxf(p.x, tb.x);
            float iy1 = fmaxf(p.y, tb.y);
            float ix2 = fminf(p.z, tb.z);
            float iy2 = fminf(p.w, tb.w);
            float inter = fmaxf(ix2 - ix1, 0.0f) * fmaxf(iy2 - iy1, 0.0f);
            float iou = inter / (a1 + sA2[t] - inter + 1e-6f);
            unsigned long long key =
                ((unsigned long long)__float_as_uint(iou) << 32) | lo;
            if (key > best[t]) best[t] = key;
        }
    }

#pragma unroll
    for (int t = 0; t < T_TGT; ++t)
        atomicMax(&sKey[t], best[t]);

    // deterministic softplus block reduction (fixed shuffle tree)
    for (int off = 16; off; off >>= 1) spsum += __shfl_xor(spsum, off, 32);
    if ((tid & 31) == 0) sRed[tid >> 5] = spsum;
    __syncthreads();

    if (tid < T_TGT) atomicMax(&wkey[b * T_TGT + tid], sKey[tid]);
    if (tid == 0) {
        float s = 0.0f;
        for (int w = 0; w < BLK / 32; ++w) s += sRed[w];
        wsp[b * BPI + blk] = s;
    }
}

// ---------------------------------------------------------------------------
// Stage 2: one wave32 per image. Dedup + sort best indices (shuffle/ballot),
// then box CIoU (TxT, reference broadcast semantics), cls BCE, obj BCE.
// ---------------------------------------------------------------------------
__global__ __launch_bounds__(32) void yolo_stage2(
    const float4* __restrict__ pboxes, const float* __restrict__ pconf,
    const float* __restrict__ pcls, const float4* __restrict__ tboxes,
    const int* __restrict__ tlabels,
    const unsigned long long* __restrict__ wkey,
    const float* __restrict__ wsp, float* __restrict__ wloss) {

    const int b = blockIdx.x;
    const int lane = threadIdx.x;

    unsigned long long key = wkey[b * T_TGT + lane];
    unsigned my = 0xFFFFFFFFu - (unsigned)(key & 0xFFFFFFFFull);

    // mark duplicates (keep first occurrence), compute rank among kept
    bool dup = false;
    for (int t = 0; t < T_TGT; ++t) {
        unsigned o = __shfl(my, t, 32);
        if (t < lane && o == my) dup = true;
    }
    bool keep = !dup;
    unsigned long long ball = __ballot(keep);
    int K = __popcll(ball);
    int rank = 0;
    for (int t = 0; t < T_TGT; ++t) {
        unsigned o = __shfl(my, t, 32);
        if (((ball >> t) & 1ull) && o < my) rank++;
    }

    __shared__ int   sP[T_TGT];
    __shared__ float sB1[T_TGT][4];
    sP[lane] = N_PREDS;       // sentinel
    __syncthreads();
    if (keep) sP[rank] = (int)my;
    __syncthreads();

    bool validi = lane < K;
    int pi = validi ? sP[lane] : 0;
    float4 b1 = pboxes[b * N_PREDS + pi];
    int ti = min(pi, T_TGT - 1);          // reference: tidx = min(pos, T-1)
    float4 b2 = tboxes[b * T_TGT + ti];
    sB1[lane][0] = b1.x; sB1[lane][1] = b1.y;
    sB1[lane][2] = b1.z; sB1[lane][3] = b1.w;
    __syncthreads();

    // per-column (j = lane) broadcast terms, exactly as reference broadcasting
    float w1 = b1.z - b1.x, h1 = b1.w - b1.y;
    float w2 = b2.z - b2.x, h2 = b2.w - b2.y;
    float a2 = w2 * h2;
    float dxe = fmaxf(b1.z, b2.z) - fminf(b1.x, b2.x);
    float dye = fmaxf(b1.w, b2.w) - fminf(b1.y, b2.y);
    float cdiag = dxe * dxe + dye * dye;
    float cx = (b1.x + b1.z) * 0.5f - (b2.x + b2.z) * 0.5f;
    float cy = (b1.y + b1.w) * 0.5f - (b2.y + b2.w) * 0.5f;
    float center = cx * cx + cy * cy;
    const float FOUR_OVER_PI2 =
        (float)(4.0 / (3.14159265358979323846 * 3.14159265358979323846));
    float dat = atanf(w2 / h2) - atanf(w1 / h1);
    float v = FOUR_OVER_PI2 * dat * dat;
    float coff = center / cdiag;

    float col = 0.0f;
    for (int i = 0; i < K; ++i) {
        float bx1 = sB1[i][0], by1 = sB1[i][1];
        float bx2 = sB1[i][2], by2 = sB1[i][3];
        float ai = (bx2 - bx1) * (by2 - by1);
        float ix1 = fmaxf(bx1, b2.x);
        float iy1 = fmaxf(by1, b2.y);
        float ix2 = fminf(bx2, b2.z);
        float iy2 = fminf(by2, b2.w);
        float inter = fmaxf(ix2 - ix1, 0.0f) * fmaxf(iy2 - iy1, 0.0f);
        float iou = inter / (ai + a2 - inter + 1e-6f);
        float alpha = v / (1.0f - iou + v + 1e-6f);
        float ciou = iou - coff - alpha * v;
        col += 1.0f - ciou;
    }
    col = validi ? col : 0.0f;
    float box_loss = wave_sum32(col) / (float)(K * K);

    // classification BCE over gathered rows
    float cacc = 0.0f;
    for (int i = 0; i < K; ++i) {
        int gi = sP[i];
        int tti = min(gi, T_TGT - 1);
        int lbl = tlabels[b * T_TGT + tti];
        const float* row = pcls + (size_t)(b * N_PREDS + gi) * C_CLS;
        for (int c = lane; c < C_CLS; c += 32) {
            float x = row[c];
            cacc += softplusf(x) - ((c == lbl) ? x : 0.0f);
        }
    }
    float cls_loss = wave_sum32(cacc) / (float)(K * C_CLS);

    // objectness: sum softplus(conf) - sum_{unique best} conf, / N
    float sub = validi ? pconf[b * N_PREDS + pi] : 0.0f;
    float sub_sum = wave_sum32(sub);
    if (lane == 0) {
        float sp = 0.0f;
        for (int w = 0; w < BPI; ++w) sp += wsp[b * BPI + w];
        float obj_loss = (sp - sub_sum) / (float)N_PREDS;
        wloss[b] = 5.0f * box_loss + obj_loss + cls_loss;
    }
}

// ---------------------------------------------------------------------------
// Stage 3: deterministic mean of the 16 per-image losses.
// ---------------------------------------------------------------------------
__global__ __launch_bounds__(32) void yolo_stage3(
    const float* __restrict__ wloss, float* __restrict__ out) {
    int lane = threadIdx.x;
    float v = (lane < B_IMGS) ? wloss[lane] : 0.0f;
    float total = wave_sum32(v) * (1.0f / (float)B_IMGS);
    if (lane == 0) out[0] = total;
}

extern "C" void kernel_launch(void* const* d_in, const int* in_sizes, int n_in,
                              void* d_out, int out_size, void* d_ws,
                              size_t ws_size, hipStream_t stream) {
    const float4* pboxes = (const float4*)d_in[0];
    const float*  pconf  = (const float*)d_in[1];
    const float*  pcls   = (const float*)d_in[2];
    const float4* tboxes = (const float4*)d_in[3];
    const int*    tlab   = (const int*)d_in[4];
    // d_in[5] (anchors) is unused by the reference computation.

    unsigned long long* wkey = (unsigned long long*)d_ws;
    float* wsp   = (float*)((char*)d_ws +
                            (size_t)B_IMGS * T_TGT * sizeof(unsigned long long));
    float* wloss = wsp + B_IMGS * BPI;

    yolo_init  <<<2, 256, 0, stream>>>(wkey);
    yolo_stage1<<<B_IMGS * BPI, BLK, 0, stream>>>(pboxes, pconf, tboxes,
                                                  wkey, wsp);
    yolo_stage2<<<B_IMGS, 32, 0, stream>>>(pboxes, pconf, pcls, tboxes, tlab,
                                           wkey, wsp, wloss);
    yolo_stage3<<<1, 32, 0, stream>>>(wloss, (float*)d_out);
}